// TaskAlignedAssigner_15564961480802
// MI455X (gfx1250) — compile-verified
//
#include <hip/hip_runtime.h>
#include <hip/hip_bf16.h>
#include <stdint.h>

#define TOPK 10
#define MAXN 8400   // N for this problem; s_key LDS array sized for it

// Order-preserving float -> u32 transform (monotone increasing).
__device__ __forceinline__ unsigned f2key(float f) {
  unsigned b = __float_as_uint(f);
  return (b & 0x80000000u) ? ~b : (b | 0x80000000u);
}

// 64-bit xor-shuffle built from two 32-bit wave32 shuffles.
__device__ __forceinline__ unsigned long long shfl_xor_u64(unsigned long long v, int m) {
  unsigned lo = __shfl_xor((unsigned)v, m);
  unsigned hi = __shfl_xor((unsigned)(v >> 32), m);
  return ((unsigned long long)hi << 32) | lo;
}

__device__ __forceinline__ float iou_clip(float4 bb, float g0, float g1, float g2, float g3,
                                          float area_g) {
  float ix1 = fmaxf(bb.x, g0), iy1 = fmaxf(bb.y, g1);
  float ix2 = fminf(bb.z, g2), iy2 = fminf(bb.w, g3);
  float inter = fmaxf(ix2 - ix1, 0.f) * fmaxf(iy2 - iy1, 0.f);
  float ap = (bb.z - bb.x) * (bb.w - bb.y);
  return fmaxf(inter / (ap + area_g - inter), 1e-9f);
}

// ---------------------------------------------------------------------------
// Phase 1: one block per batch. Metric + fallback + block top-10 -> workspace.
// ---------------------------------------------------------------------------
__global__ __launch_bounds__(256)
void tal_phase1(const float* __restrict__ scores,   // (B,N,C)
                const float* __restrict__ boxes,    // (B,N,4)
                const float* __restrict__ anchors,  // (N,2)
                const int*   __restrict__ labels,   // (B,)
                const float* __restrict__ gts,      // (B,1,4)
                int N, int C,
                int* __restrict__ wsIdx, float* __restrict__ wsVal,
                int* __restrict__ wsSel)
{
  __shared__ unsigned s_key[MAXN];
  __shared__ unsigned long long s_red[8];
  __shared__ int s_any;
  __shared__ unsigned long long s_fb;
  __shared__ int s_selIdx[TOPK];
  __shared__ unsigned s_selKey[TOPK];

  const int b    = blockIdx.x;
  const int tid  = threadIdx.x;
  const int wid  = tid >> 5;
  const int lane = tid & 31;
  const int nw   = (int)(blockDim.x >> 5);

  const float g0 = gts[b * 4 + 0], g1 = gts[b * 4 + 1];
  const float g2 = gts[b * 4 + 2], g3 = gts[b * 4 + 3];
  const int   cls = labels[b];
  const bool  gt_valid = (g2 > g0) && (g3 > g1);
  const float gcx = (g0 + g2) * 0.5f, gcy = (g1 + g3) * 0.5f;
  const float area_g = (g2 - g0) * (g3 - g1);

  if (tid == 0) { s_any = 0; s_fb = ~0ull; }
  __syncthreads();

  const float4* boxb = (const float4*)boxes + (size_t)b * N;
  const float*  scb  = scores + (size_t)b * N * C + cls;  // strided column gather

  int any_loc = 0;
  unsigned long long fb_loc = ~0ull;
  const unsigned km1 = f2key(-1.0f);

  for (int i = tid; i < N; i += blockDim.x) {
    __builtin_prefetch(&scb[(size_t)(i + blockDim.x) * C], 0, 0);  // global_prefetch_b8
    float cx = anchors[2 * i], cy = anchors[2 * i + 1];
    bool in = (cx >= g0) && (cx <= g2) && (cy >= g1) && (cy <= g3);
    float dx = cx - gcx, dy = cy - gcy;
    float d2 = dx * dx + dy * dy;  // >= 0, bit pattern is order-preserving
    unsigned long long p = ((unsigned long long)__float_as_uint(d2) << 32) | (unsigned)i;
    fb_loc = (p < fb_loc) ? p : fb_loc;   // min d2, tie -> lower index (argmin semantics)
    any_loc |= (int)in;

    float iou = iou_clip(boxb[i], g0, g1, g2, g3, area_g);
    float s   = scb[(size_t)i * C];
    float sig = 1.f / (1.f + __expf(-s));
    float p2  = iou * iou;
    float metric = sqrtf(sig) * (p2 * p2 * p2);   // cls^0.5 * iou^6, >= 0
    s_key[i] = in ? f2key(metric) : km1;
  }

  // Block reduce any_in and fallback argmin.
  for (int m = 16; m > 0; m >>= 1) {
    unsigned long long o = shfl_xor_u64(fb_loc, m);
    fb_loc = (o < fb_loc) ? o : fb_loc;
    any_loc |= __shfl_xor(any_loc, m);
  }
  if (lane == 0) {
    s_red[wid] = fb_loc;
    if (any_loc) s_any = 1;  // benign race: all writers store 1
  }
  __syncthreads();
  if (tid == 0) {
    unsigned long long m = s_red[0];
    for (int w = 1; w < nw; w++) m = (s_red[w] < m) ? s_red[w] : m;
    s_fb = m;
  }
  __syncthreads();

  if (tid == 0 && !s_any) {
    int fi = (int)(unsigned)s_fb;  // fallback: nearest-center anchor becomes valid
    float iou = iou_clip(boxb[fi], g0, g1, g2, g3, area_g);
    float s   = scb[(size_t)fi * C];
    float sig = 1.f / (1.f + __expf(-s));
    float p2  = iou * iou;
    s_key[fi] = f2key(sqrtf(sig) * (p2 * p2 * p2));
  }
  __syncthreads();

  // Top-10: 10 rounds of block arg-max with top_k tie-break (lower index first).
  for (int k = 0; k < TOPK; k++) {
    unsigned long long best = 0;
    for (int i = tid; i < N; i += blockDim.x) {
      unsigned long long p = ((unsigned long long)s_key[i] << 32) | (unsigned)(N - i);
      best = (p > best) ? p : best;
    }
    for (int m = 16; m > 0; m >>= 1) {
      unsigned long long o = shfl_xor_u64(best, m);
      best = (o > best) ? o : best;
    }
    if (lane == 0) s_red[wid] = best;
    __syncthreads();
    if (tid == 0) {
      unsigned long long m = s_red[0];
      for (int w = 1; w < nw; w++) m = (s_red[w] > m) ? s_red[w] : m;
      int idx = N - (int)(unsigned)m;
      s_selIdx[k] = idx;
      s_selKey[k] = (unsigned)(m >> 32);
      s_key[idx]  = 0u;  // knock out winner
    }
    __syncthreads();
  }

  if (tid < TOPK) {
    int idx = s_selIdx[tid];
    bool sel = (s_selKey[tid] >= 0x80000000u) && gt_valid;  // metric >= 0 && gt_valid
    float val = 0.f;
    if (sel) val = iou_clip(boxb[idx], g0, g1, g2, g3, area_g);
    wsIdx[b * TOPK + tid] = idx;
    wsVal[b * TOPK + tid] = val;
    wsSel[b * TOPK + tid] = sel ? 1 : 0;
  }
}

// ---------------------------------------------------------------------------
// Phase 2: zero-fill all outputs via CDNA5 async LDS->global store engine.
// LDS tile of zeros is written once; each lane streams b128 async stores
// (ASYNCcnt-tracked), drained with s_wait_asynccnt before kernel end.
// ---------------------------------------------------------------------------
__global__ __launch_bounds__(256)
void tal_zero_async(float4* __restrict__ out4, long long n4,
                    float* __restrict__ tail, int ntail)
{
  __shared__ float4 zbuf[256];
  const int tid = threadIdx.x;
  zbuf[tid] = make_float4(0.f, 0.f, 0.f, 0.f);
  __syncthreads();  // DS writes complete before the async engine reads LDS

  // Low 32 bits of the generic pointer to an LDS object == LDS byte offset
  // (flat addressing); VSRC must be ONE VGPR.
  const unsigned lds_off = (unsigned)(uintptr_t)&zbuf[tid];

  long long i = (long long)blockIdx.x * blockDim.x + tid;
  const long long stride = (long long)gridDim.x * blockDim.x;
  for (; i < n4; i += stride) {
    unsigned long long gaddr = (unsigned long long)(uintptr_t)(out4 + i);
    asm volatile("global_store_async_from_lds_b128 %0, %1, off"
                 :: "v"(gaddr), "v"(lds_off)
                 : "memory");
  }
  if (blockIdx.x == 0 && tid < ntail) tail[tid] = 0.f;
  asm volatile("s_wait_asynccnt 0" ::: "memory");
}

// ---------------------------------------------------------------------------
// Phase 3: splice the <=10 nonzero entries per batch into tb / ts / fg.
// ---------------------------------------------------------------------------
__global__ void tal_fixup(const int* __restrict__ wsIdx, const float* __restrict__ wsVal,
                          const int* __restrict__ wsSel,
                          const float* __restrict__ gts, const int* __restrict__ labels,
                          float* __restrict__ out, int N, int C)
{
  const int b = blockIdx.x;
  const int k = threadIdx.x;
  if (k >= TOPK) return;
  if (!wsSel[b * TOPK + k]) return;
  const int idx = wsIdx[b * TOPK + k];
  const float v = wsVal[b * TOPK + k];
  const size_t bn = (size_t)b * N + idx;
  const size_t B  = gridDim.x;

  float4 g = *(const float4*)(gts + (size_t)b * 4);
  ((float4*)out)[bn] = g;                               // tb row (16B aligned)

  const size_t tsBase = B * (size_t)N * 4;
  out[tsBase + bn * C + labels[b]] = v;                 // ts[idx, cls] = iou

  const size_t fgBase = tsBase + B * (size_t)N * C;
  out[fgBase + bn] = 1.0f;                              // fg[idx] = true
}

extern "C" void kernel_launch(void* const* d_in, const int* in_sizes, int n_in,
                              void* d_out, int out_size, void* d_ws, size_t ws_size,
                              hipStream_t stream)
{
  const float* scores  = (const float*)d_in[0];   // pred_scores (B,N,C)
  const float* boxes   = (const float*)d_in[1];   // pred_bboxes_xyxy (B,N,4)
  const float* anchors = (const float*)d_in[2];   // anchors (N,2)
  const int*   labels  = (const int*)  d_in[3];   // gt_labels (B,)
  const float* gts     = (const float*)d_in[4];   // gt_bboxes_xyxy (B,1,4)
  float* out = (float*)d_out;

  const int B = in_sizes[3];
  const int N = in_sizes[2] / 2;
  const int C = in_sizes[0] / (B * N);

  int*   wsIdx = (int*)d_ws;
  float* wsVal = (float*)((char*)d_ws + (size_t)B * TOPK * 4);
  int*   wsSel = (int*)((char*)d_ws + (size_t)B * TOPK * 8);

  tal_phase1<<<B, 256, 0, stream>>>(scores, boxes, anchors, labels, gts,
                                    N, C, wsIdx, wsVal, wsSel);

  const long long n4 = (long long)out_size / 4;
  const int ntail = out_size & 3;
  float* tail = out + (size_t)n4 * 4;
  tal_zero_async<<<2048, 256, 0, stream>>>((float4*)out, n4, tail, ntail);

  tal_fixup<<<B, 32, 0, stream>>>(wsIdx, wsVal, wsSel, gts, labels, out, N, C);
}